// TSNE2D_23287312679476
// MI455X (gfx1250) — compile-verified
//
#include <hip/hip_runtime.h>
#include <hip/hip_bf16.h>
#include <math.h>

// t-SNE (N=4096, d=64, 2-D embed, 300 iters) for gfx1250 / MI455X.
// Workspace layout (requires ~134.6 MB):
//   [0,            NN*4)      D  (squared dists), later reused as Pu (exp terms)
//   [NN*4,         2*NN*4)    P  (final joint probabilities)
//   [2*NN*4, ...)             small arrays (norms, beta, rowsums, upd, gains,
//                             grad, block partials, scalars)

typedef float v2f __attribute__((ext_vector_type(2)));
typedef float v8f __attribute__((ext_vector_type(8)));

#define TSNE_N    4096
#define TSNE_DIM  64
#define TSNE_NN   ((size_t)TSNE_N * (size_t)TSNE_N)
#define TSNE_EPS  1e-12f
#define TSNE_LR   200.0f

// ---------------------------------------------------------------- row norms
__global__ void tsne_norms(const float* __restrict__ X, float* __restrict__ s) {
  int i = blockIdx.x * blockDim.x + threadIdx.x;
  if (i >= TSNE_N) return;
  const float* row = X + (size_t)i * TSNE_DIM;
  float acc = 0.0f;
#pragma unroll
  for (int k = 0; k < TSNE_DIM; ++k) acc += row[k] * row[k];
  s[i] = acc;
}

// ------------------------------------------- D = ||xi||^2+||xj||^2-2 X X^T
// One wave32 per 16x16 tile; K=64 via 16 x V_WMMA_F32_16X16X4_F32.
__global__ __launch_bounds__(32) void tsne_gram_dist(
    const float* __restrict__ X, const float* __restrict__ s,
    float* __restrict__ D) {
  const int tile = blockIdx.x;             // 256*256 tiles
  const int ti   = (tile >> 8) << 4;       // row block base
  const int tj   = (tile & 255) << 4;      // col block base
  const int lane = threadIdx.x;            // 0..31 (wave32)
  const int half = lane >> 4;              // selects K pair
  const int l15  = lane & 15;

  // A per lane: X[ti+l15][k0 + 2*half .. +1]   (16x4 f32 operand)
  // B per lane: X[tj+l15][k0 + 2*half .. +1]   (4x16 f32 operand, B[k][n]=X[tj+n][k])
  const float* arow = X + (size_t)(ti + l15) * TSNE_DIM + half * 2;
  const float* brow = X + (size_t)(tj + l15) * TSNE_DIM + half * 2;

  v8f acc = {};
#pragma unroll
  for (int k0 = 0; k0 < TSNE_DIM; k0 += 4) {
    v2f a = { arow[k0], arow[k0 + 1] };
    v2f b = { brow[k0], brow[k0 + 1] };
    acc = __builtin_amdgcn_wmma_f32_16x16x4_f32(
        /*neg_a=*/false, a, /*neg_b=*/false, b,
        /*c_mod=*/(short)0, acc, /*reuse_a=*/false, /*reuse_b=*/false);
  }

  const float sj = s[tj + l15];
#pragma unroll
  for (int v = 0; v < 8; ++v) {
    const int m  = v + (half << 3);          // C/D layout: lanes 16-31 -> M+8
    const int gi = ti + m;
    const int gj = tj + l15;
    float d = s[gi] + sj - 2.0f * acc[v];
    D[(size_t)gi * TSNE_N + gj] = fmaxf(d, 0.0f);
  }
}

// ------------------------------------------------------- per-row bisection
// One block (256 threads) per row; 16 distances held in registers per lane.
__global__ __launch_bounds__(256) void tsne_beta(
    const float* __restrict__ D, float* __restrict__ beta_out) {
  const int i = blockIdx.x;
  const int t = threadIdx.x;
  float d[16], m[16];
#pragma unroll
  for (int u = 0; u < 16; ++u) {
    int j = t + (u << 8);
    d[u] = D[(size_t)i * TSNE_N + j];
    m[u] = (j == i) ? 0.0f : 1.0f;
  }
  __shared__ float redP[256];
  __shared__ float redD[256];
  __shared__ float sBeta, sLo, sHi;
  if (t == 0) { sBeta = 1.0f; sLo = 0.0f; sHi = __builtin_huge_valf(); }
  __syncthreads();

  const float logU = logf(50.0f);   // log(perplexity)
  for (int step = 0; step < 50; ++step) {
    const float beta = sBeta;
    float sp = 0.0f, sdp = 0.0f;
#pragma unroll
    for (int u = 0; u < 16; ++u) {
      float p = __expf(-d[u] * beta) * m[u];
      sp  += p;
      sdp += d[u] * p;
    }
    redP[t] = sp; redD[t] = sdp;
    __syncthreads();
    for (int off = 128; off > 0; off >>= 1) {
      if (t < off) { redP[t] += redP[t + off]; redD[t] += redD[t + off]; }
      __syncthreads();
    }
    if (t == 0) {
      float sumP = fmaxf(redP[0], TSNE_EPS);
      float H    = logf(sumP) + beta * redD[0] / sumP;
      bool  too_high = H > logU;
      float lo = sLo, hi = sHi;
      if (too_high) lo = beta; else hi = beta;
      float nb = too_high ? (__builtin_isinf(hi) ? beta * 2.0f
                                                 : 0.5f * (beta + hi))
                          : 0.5f * (beta + lo);
      sLo = lo; sHi = hi; sBeta = nb;
    }
    __syncthreads();
  }
  if (t == 0) beta_out[i] = sBeta;
}

// ------------------------------- Pu = exp(-D*beta_i)*mask (in place), rowsum
__global__ __launch_bounds__(256) void tsne_pu(
    float* __restrict__ DPu, const float* __restrict__ beta,
    float* __restrict__ rsum) {
  const int i = blockIdx.x;
  const int t = threadIdx.x;
  const float b = beta[i];
  float acc = 0.0f;
  for (int j = t; j < TSNE_N; j += 256) {
    size_t idx = (size_t)i * TSNE_N + j;
    float p = (j == i) ? 0.0f : __expf(-DPu[idx] * b);
    DPu[idx] = p;
    acc += p;
  }
  __shared__ float red[256];
  red[t] = acc;
  __syncthreads();
  for (int off = 128; off > 0; off >>= 1) {
    if (t < off) red[t] += red[t + off];
    __syncthreads();
  }
  if (t == 0) rsum[i] = red[0];
}

// ---------------------- P = Pn + Pn^T, with per-block partial of total sum
__global__ __launch_bounds__(256) void tsne_sym(
    const float* __restrict__ Pu, const float* __restrict__ rsum,
    float* __restrict__ P, float* __restrict__ partials) {
  const size_t idx = (size_t)blockIdx.x * 256 + threadIdx.x;
  const int i = (int)(idx >> 12);
  const int j = (int)(idx & 4095);
  float pij = Pu[idx] / fmaxf(rsum[i], TSNE_EPS);
  float pji = Pu[(size_t)j * TSNE_N + i] / fmaxf(rsum[j], TSNE_EPS);
  float v = pij + pji;
  P[idx] = v;
  __shared__ float red[256];
  red[threadIdx.x] = v;
  __syncthreads();
  for (int off = 128; off > 0; off >>= 1) {
    if (threadIdx.x < off) red[threadIdx.x] += red[threadIdx.x + off];
    __syncthreads();
  }
  if (threadIdx.x == 0) partials[blockIdx.x] = red[0];
}

// -------------------------------------------- generic deterministic reduce
__global__ __launch_bounds__(256) void tsne_reduce(
    const float* __restrict__ in, int n, float* __restrict__ out) {
  const int t = threadIdx.x;
  float acc = 0.0f;
  for (int idx = t; idx < n; idx += 256) acc += in[idx];
  __shared__ float red[256];
  red[t] = acc;
  __syncthreads();
  for (int off = 128; off > 0; off >>= 1) {
    if (t < off) red[t] += red[t + off];
    __syncthreads();
  }
  if (t == 0) out[0] = red[0];
}

// ---------------------------------------- P = max(P / max(T, eps), eps)
__global__ __launch_bounds__(256) void tsne_scale(
    float* __restrict__ P, const float* __restrict__ T) {
  const size_t idx = (size_t)blockIdx.x * 256 + threadIdx.x;
  const float invT = 1.0f / fmaxf(T[0], TSNE_EPS);
  P[idx] = fmaxf(P[idx] * invT, TSNE_EPS);
}

// ------------------------------------------------------------ init state
__global__ void tsne_init(const float* __restrict__ y0, float* __restrict__ Y,
                          float* __restrict__ upd, float* __restrict__ gains) {
  int idx = blockIdx.x * blockDim.x + threadIdx.x;
  if (idx >= TSNE_N * 2) return;
  Y[idx]     = y0[idx];
  upd[idx]   = 0.0f;
  gains[idx] = 1.0f;
}

// ---------------------- per-row partial of Z = sum_{i!=j} 1/(1+||yi-yj||^2)
__global__ __launch_bounds__(256) void tsne_qsum(
    const float* __restrict__ Y, float* __restrict__ partials) {
  const int i = blockIdx.x;
  const int t = threadIdx.x;
  const float yi0 = Y[2 * i], yi1 = Y[2 * i + 1];
  float acc = 0.0f;
  for (int j = t; j < TSNE_N; j += 256) {
    float dx = yi0 - Y[2 * j];
    float dy = yi1 - Y[2 * j + 1];
    float num = (j == i) ? 0.0f : 1.0f / (1.0f + dx * dx + dy * dy);
    acc += num;
  }
  __shared__ float red[256];
  red[t] = acc;
  __syncthreads();
  for (int off = 128; off > 0; off >>= 1) {
    if (t < off) red[t] += red[t + off];
    __syncthreads();
  }
  if (t == 0) partials[i] = red[0];
}

// ------------------------------------------------------------- gradient
__global__ __launch_bounds__(256) void tsne_grad(
    const float* __restrict__ Y, const float* __restrict__ P,
    const float* __restrict__ Ssum, float exag, float* __restrict__ grad) {
  const int i = blockIdx.x;
  const int t = threadIdx.x;
  const float invS = 1.0f / fmaxf(Ssum[0], TSNE_EPS);
  const float yi0 = Y[2 * i], yi1 = Y[2 * i + 1];
  const float* __restrict__ Prow = P + (size_t)i * TSNE_N;
  float g0 = 0.0f, g1 = 0.0f;
  for (int j = t; j < TSNE_N; j += 256) {
    __builtin_prefetch(&Prow[j + 512], 0, 1);   // global_prefetch_b8
    float dx = yi0 - Y[2 * j];
    float dy = yi1 - Y[2 * j + 1];
    float num = (j == i) ? 0.0f : 1.0f / (1.0f + dx * dx + dy * dy);
    float q  = fmaxf(num * invS, TSNE_EPS);
    float pq = (exag * Prow[j] - q) * num;
    g0 += pq * dx;
    g1 += pq * dy;
  }
  __shared__ float red0[256];
  __shared__ float red1[256];
  red0[t] = g0; red1[t] = g1;
  __syncthreads();
  for (int off = 128; off > 0; off >>= 1) {
    if (t < off) { red0[t] += red0[t + off]; red1[t] += red1[t + off]; }
    __syncthreads();
  }
  if (t == 0) {
    grad[2 * i]     = 4.0f * red0[0];
    grad[2 * i + 1] = 4.0f * red1[0];
  }
}

// ------------------------------------------------- gains / momentum update
__global__ void tsne_update(float* __restrict__ Y, float* __restrict__ upd,
                            float* __restrict__ gains,
                            const float* __restrict__ grad, float momentum) {
  int idx = blockIdx.x * blockDim.x + threadIdx.x;
  if (idx >= TSNE_N * 2) return;
  float g  = grad[idx];
  float u  = upd[idx];
  float gn = gains[idx];
  bool inc = (g * u) < 0.0f;
  gn = fmaxf(inc ? (gn + 0.2f) : (gn * 0.8f), 0.01f);
  u  = momentum * u - TSNE_LR * (gn * g);
  Y[idx]    += u;
  upd[idx]   = u;
  gains[idx] = gn;
}

// =========================================================================
extern "C" void kernel_launch(void* const* d_in, const int* in_sizes, int n_in,
                              void* d_out, int out_size, void* d_ws, size_t ws_size,
                              hipStream_t stream) {
  (void)in_sizes; (void)n_in; (void)out_size; (void)ws_size;
  const float* x  = (const float*)d_in[0];   // [4096, 64]
  const float* y0 = (const float*)d_in[1];   // [4096, 2]
  float* Y = (float*)d_out;                  // embedding lives in d_out

  char*  ws  = (char*)d_ws;
  float* D   = (float*)ws;                          // NN floats (later Pu)
  float* P   = (float*)(ws + TSNE_NN * 4);          // NN floats
  char*  sm  = ws + 2 * TSNE_NN * 4;
  float* s        = (float*)sm;  sm += TSNE_N * 4;
  float* beta     = (float*)sm;  sm += TSNE_N * 4;
  float* rsum     = (float*)sm;  sm += TSNE_N * 4;
  float* upd      = (float*)sm;  sm += TSNE_N * 2 * 4;
  float* gains    = (float*)sm;  sm += TSNE_N * 2 * 4;
  float* grad     = (float*)sm;  sm += TSNE_N * 2 * 4;
  float* partials = (float*)sm;  sm += 65536 * 4;
  float* Tsum     = (float*)sm;  sm += 4;
  float* Ssum     = (float*)sm;  sm += 4;

  const int nTiles = (TSNE_N / 16) * (TSNE_N / 16);       // 65536
  const int nElemB = (int)(TSNE_NN / 256);                // 65536

  // ---- Phase A: squared distance matrix via WMMA Gram matrix
  tsne_norms<<<TSNE_N / 256, 256, 0, stream>>>(x, s);
  tsne_gram_dist<<<nTiles, 32, 0, stream>>>(x, s, D);

  // ---- Phase B: per-row precision search
  tsne_beta<<<TSNE_N, 256, 0, stream>>>(D, beta);

  // ---- Phase C: joint probabilities
  tsne_pu<<<TSNE_N, 256, 0, stream>>>(D, beta, rsum);     // D -> Pu in place
  tsne_sym<<<nElemB, 256, 0, stream>>>(D, rsum, P, partials);
  tsne_reduce<<<1, 256, 0, stream>>>(partials, nElemB, Tsum);
  tsne_scale<<<nElemB, 256, 0, stream>>>(P, Tsum);

  // ---- Phase D: 300 gradient-descent iterations
  tsne_init<<<(TSNE_N * 2 + 255) / 256, 256, 0, stream>>>(y0, Y, upd, gains);
  for (int t = 0; t < 300; ++t) {
    const float momentum = (t < 250) ? 0.5f : 0.8f;
    const float exag     = (t < 250) ? 12.0f : 1.0f;
    tsne_qsum<<<TSNE_N, 256, 0, stream>>>(Y, partials);
    tsne_reduce<<<1, 256, 0, stream>>>(partials, TSNE_N, Ssum);
    tsne_grad<<<TSNE_N, 256, 0, stream>>>(Y, P, Ssum, exag, grad);
    tsne_update<<<(TSNE_N * 2 + 255) / 256, 256, 0, stream>>>(Y, upd, gains, grad, momentum);
  }
}